// FractalDimensionModule_28819230556212
// MI455X (gfx1250) — compile-verified
//
#include <hip/hip_runtime.h>

// MI455X / gfx1250: wave32, WMMA (no MFMA).
typedef __attribute__((ext_vector_type(2))) float v2f;
typedef __attribute__((ext_vector_type(8))) float v8f;

#define LN2F 0.69314718055994530942f

// ---------------- Kernel 0: zero the 13 (padded to 16) per-level sum accumulators ----------------
__global__ void fd_zero(float* __restrict__ sums) {
    if (threadIdx.x < 16) sums[threadIdx.x] = 0.0f;
}

// ---------------- Kernel 1: levels 1..6, one 64x64 tile per block ----------------
// Grid: 16384 blocks (128x128 tiles), 256 threads (8 wave32).
// Each thread owns a 4x4 patch -> level1 (4 diffs) + level2 (1 diff) in registers.
// Wave-level sum reduction done with V_WMMA_F32_16X16X4_F32 (A = ones) + 4 shuffle steps.
// Levels 3..6 via LDS pyramid. Emits per-tile level-6 (max,min) for stage 2.
__global__ __launch_bounds__(256) void fd_stage1(const float* __restrict__ X,
                                                 float* __restrict__ sums,
                                                 float* __restrict__ mx6,
                                                 float* __restrict__ mn6) {
    const int bid = blockIdx.x;
    const int bx = bid & 127;          // tile x (of 128)
    const int by = bid >> 7;           // tile y (of 128)
    const int tid = (int)threadIdx.x;  // 0..255
    const int tx = tid & 15;           // 16x16 thread grid
    const int ty = tid >> 4;

    // LDS pyramid: 16x16 (256) + 8x8 (64) + 4x4 (16) + 2x2 (4) + 1 = 341 each
    __shared__ float smx[341];
    __shared__ float smn[341];
    __shared__ float lsum[6];
    if (tid < 6) lsum[tid] = 0.0f;

    const size_t W = 8192;
    const float* base = X + (size_t)(by * 64 + ty * 4) * W + (size_t)(bx * 64 + tx * 4);
    // 16 consecutive lanes read 256 contiguous bytes per row -> coalesced b128 loads
    float4 r0 = *reinterpret_cast<const float4*>(base);
    float4 r1 = *reinterpret_cast<const float4*>(base + W);
    float4 r2 = *reinterpret_cast<const float4*>(base + 2 * W);
    float4 r3 = *reinterpret_cast<const float4*>(base + 3 * W);

    // ---- level 1: four 2x2 pools inside the 4x4 patch ----
    float mxa = fmaxf(fmaxf(r0.x, r0.y), fmaxf(r1.x, r1.y));
    float mna = fminf(fminf(r0.x, r0.y), fminf(r1.x, r1.y));
    float mxb = fmaxf(fmaxf(r0.z, r0.w), fmaxf(r1.z, r1.w));
    float mnb = fminf(fminf(r0.z, r0.w), fminf(r1.z, r1.w));
    float mxc = fmaxf(fmaxf(r2.x, r2.y), fmaxf(r3.x, r3.y));
    float mnc = fminf(fminf(r2.x, r2.y), fminf(r3.x, r3.y));
    float mxd = fmaxf(fmaxf(r2.z, r2.w), fmaxf(r3.z, r3.w));
    float mnd = fminf(fminf(r2.z, r2.w), fminf(r3.z, r3.w));
    float s1 = (mxa - mna) + (mxb - mnb) + (mxc - mnc) + (mxd - mnd);

    // ---- level 2: pool the four level-1 results ----
    float mx2 = fmaxf(fmaxf(mxa, mxb), fmaxf(mxc, mxd));
    float mn2 = fminf(fminf(mna, mnb), fminf(mnc, mnd));
    float s2 = mx2 - mn2;

    smx[ty * 16 + tx] = mx2;
    smn[ty * 16 + tx] = mn2;

    // ---- WMMA wave reduction: D = ones(16x4) x B + C => row 0 of D holds column sums of B.
    // B carries one per-lane value (second K-slot zero); every lane value appears exactly
    // once in B, so sum over the 16 columns of row 0 == exact wave total. EXEC is all-1s here.
    v2f aOnes = {1.0f, 1.0f};
    v2f b1 = {s1, 0.0f};
    v2f b2 = {s2, 0.0f};
    v8f c1 = {0.0f, 0.0f, 0.0f, 0.0f, 0.0f, 0.0f, 0.0f, 0.0f};
    v8f c2 = {0.0f, 0.0f, 0.0f, 0.0f, 0.0f, 0.0f, 0.0f, 0.0f};
    c1 = __builtin_amdgcn_wmma_f32_16x16x4_f32(false, aOnes, false, b1, (short)0, c1, false, false);
    c2 = __builtin_amdgcn_wmma_f32_16x16x4_f32(false, aOnes, false, b2, (short)0, c2, false, false);
    float w1 = c1[0];  // lane n<16: column-sum n
    float w2 = c2[0];
#pragma unroll
    for (int off = 1; off <= 8; off <<= 1) {  // sum columns 0..15 (within 16-lane halves)
        w1 += __shfl_xor(w1, off);
        w2 += __shfl_xor(w2, off);
    }

    __syncthreads();  // lsum init + level-2 LDS stores visible
    if ((tid & 31) == 0) {
        atomicAdd(&lsum[0], w1);
        atomicAdd(&lsum[1], w2);
    }

    // ---- levels 3..6 in LDS ----
    int inOff = 0, inW = 16, outOff = 256;
#pragma unroll
    for (int l = 2; l <= 5; ++l) {
        __syncthreads();
        int outW = inW >> 1;
        if (tid < outW * outW) {
            int x = tid % outW;
            int y = tid / outW;
            int i00 = inOff + (2 * y) * inW + 2 * x;
            int i10 = i00 + inW;
            float mv = fmaxf(fmaxf(smx[i00], smx[i00 + 1]), fmaxf(smx[i10], smx[i10 + 1]));
            float nv = fminf(fminf(smn[i00], smn[i00 + 1]), fminf(smn[i10], smn[i10 + 1]));
            atomicAdd(&lsum[l], mv - nv);
            smx[outOff + y * outW + x] = mv;
            smn[outOff + y * outW + x] = nv;
        }
        inOff = outOff;
        outOff += outW * outW;
        inW = outW;
    }
    __syncthreads();
    if (tid == 0) {
        mx6[bid] = smx[340];
        mn6[bid] = smn[340];
    }
    if (tid < 6) atomicAdd(&sums[tid], lsum[tid]);
}

// ---------------- Kernel 2: levels 7..13 from the 128x128 residual (single block) ----------------
__global__ __launch_bounds__(1024) void fd_stage2(const float* __restrict__ mx6,
                                                  const float* __restrict__ mn6,
                                                  float* __restrict__ sums) {
    const int tid = (int)threadIdx.x;
    // 64^2 + 32^2 + 16^2 + 8^2 + 4^2 + 2^2 + 1 = 5461 each (~44 KB total LDS)
    __shared__ float smx[5461];
    __shared__ float smn[5461];
    __shared__ float lsum[7];
    if (tid < 7) lsum[tid] = 0.0f;
    __syncthreads();

    // ---- level 7: 128x128 -> 64x64 (from global) ----
    float acc = 0.0f;
    for (int o = tid; o < 4096; o += 1024) {
        int x = o & 63, y = o >> 6;
        int g00 = (2 * y) * 128 + 2 * x;
        int g10 = g00 + 128;
        float mv = fmaxf(fmaxf(mx6[g00], mx6[g00 + 1]), fmaxf(mx6[g10], mx6[g10 + 1]));
        float nv = fminf(fminf(mn6[g00], mn6[g00 + 1]), fminf(mn6[g10], mn6[g10 + 1]));
        smx[o] = mv;
        smn[o] = nv;
        acc += mv - nv;
    }
#pragma unroll
    for (int off = 1; off < 32; off <<= 1) acc += __shfl_xor(acc, off);
    if ((tid & 31) == 0) atomicAdd(&lsum[0], acc);

    // ---- levels 8..13 in LDS ----
    int inOff = 0, inW = 64, outOff = 4096;
    for (int l = 1; l <= 6; ++l) {
        __syncthreads();
        int outW = inW >> 1;
        if (tid < outW * outW) {
            int x = tid % outW;
            int y = tid / outW;
            int i00 = inOff + (2 * y) * inW + 2 * x;
            int i10 = i00 + inW;
            float mv = fmaxf(fmaxf(smx[i00], smx[i00 + 1]), fmaxf(smx[i10], smx[i10 + 1]));
            float nv = fminf(fminf(smn[i00], smn[i00 + 1]), fminf(smn[i10], smn[i10 + 1]));
            atomicAdd(&lsum[l], mv - nv);
            smx[outOff + y * outW + x] = mv;
            smn[outOff + y * outW + x] = nv;
        }
        inOff = outOff;
        outOff += outW * outW;
        inW = outW;
    }
    __syncthreads();
    if (tid < 7) atomicAdd(&sums[6 + tid], lsum[tid]);
}

// ---------------- Kernel 3: closed-form 13-point log-log least squares ----------------
// sums[k] = sum of (max-min) at pooling size 2^(k+1), array count N_k = 4^(12-k).
// mean = sum / N_k, so log(mean) = log(sum) - 2*(12-k)*ln2. Slope fit, return -m.
__global__ void fd_fit(const float* __restrict__ sums, float* __restrict__ out) {
    if (threadIdx.x != 0 || blockIdx.x != 0) return;
    float Sx = 0.0f, Sxx = 0.0f, Syb = 0.0f, Sxyb = 0.0f, Sym = 0.0f, Sxym = 0.0f;
    for (int k = 0; k < 13; ++k) {
        float x = (float)(k + 1) * LN2F;                  // log(size)
        float yb = logf(sums[k]);                         // log(boxcount sum)
        float ym = yb - 2.0f * (float)(12 - k) * LN2F;    // log(mean)
        Sx += x;
        Sxx += x * x;
        Syb += yb;
        Sxyb += x * yb;
        Sym += ym;
        Sxym += x * ym;
    }
    const float L = 13.0f;
    float denom = L * Sxx - Sx * Sx;
    out[0] = -(L * Sxyb - Sx * Syb) / denom;  // fd_b
    out[1] = -(L * Sxym - Sx * Sym) / denom;  // fd_m
}

extern "C" void kernel_launch(void* const* d_in, const int* in_sizes, int n_in,
                              void* d_out, int out_size, void* d_ws, size_t ws_size,
                              hipStream_t stream) {
    const float* X = (const float*)d_in[0];     // 8192*8192 fp32
    float* out = (float*)d_out;                 // 2 fp32
    char* ws = (char*)d_ws;
    float* sums = (float*)ws;                        // 16 floats (13 used)
    float* mx6 = (float*)(ws + 256);                 // 128*128 floats
    float* mn6 = (float*)(ws + 256 + 128 * 128 * 4); // 128*128 floats

    fd_zero<<<1, 32, 0, stream>>>(sums);
    fd_stage1<<<16384, 256, 0, stream>>>(X, sums, mx6, mn6);
    fd_stage2<<<1, 1024, 0, stream>>>(mx6, mn6, sums);
    fd_fit<<<1, 32, 0, stream>>>(sums, out);
}